// DATMambaLayer_16965120820096
// MI455X (gfx1250) — compile-verified
//
#include <hip/hip_runtime.h>
#include <math.h>

typedef float v2f __attribute__((ext_vector_type(2)));
typedef float v8f __attribute__((ext_vector_type(8)));

#define LN_EPS 1e-5f

__device__ __forceinline__ float silu_f(float x) { return x / (1.0f + expf(-x)); }
__device__ __forceinline__ float gelu_f(float x) { return 0.5f * x * (1.0f + erff(x * 0.70710678118654752f)); }
__device__ __forceinline__ float softplus_f(float x) { return x > 20.0f ? x : log1pf(expf(x)); }
__device__ __forceinline__ float sigmoid_f(float x) { return 1.0f / (1.0f + expf(-x)); }
__device__ __forceinline__ int   imin(int a, int b) { return a < b ? a : b; }

// ---------------------------------------------------------------------------
// Generic WMMA fp32 GEMM:  C[M,N] = act(A[M,K] * W[N,K]^T + bias)
// A row-major (lda), W row-major weight layout (ldw = K stride), C row-major.
// Wave computes a 16x64 strip with V_WMMA_F32_16X16X4_F32; block = 8 waves
// covering 128 rows x 64 cols.  act: 0=none 1=silu 2=gelu 3=softplus
//
// Out-of-range rows/cols: load addresses are CLAMPED to the last valid
// row/column (never predicated) -- each WMMA output element (m,n) depends only
// on A row m and B column n, and out-of-range outputs are masked at the store,
// so clamping is exact and keeps the inner loop branch-free / EXEC-stable.
// ---------------------------------------------------------------------------
__global__ __launch_bounds__(256) void k_gemm(
    const float* __restrict__ A, int lda,
    const float* __restrict__ W, int ldw,
    float* __restrict__ C, int ldc,
    int M, int N, int K,
    const float* __restrict__ bias, int act)
{
  const int lane = threadIdx.x & 31;
  const int wave = threadIdx.x >> 5;
  const int ln   = lane & 15;
  const int kh   = lane >> 4;                 // 0/1 : which K-pair this lane holds
  const int row0 = (blockIdx.y * 8 + wave) * 16;
  const int col0 = blockIdx.x * 64;

  const v8f zero8 = {0.f,0.f,0.f,0.f,0.f,0.f,0.f,0.f};
  v8f acc[4] = {zero8, zero8, zero8, zero8};

  const int am = imin(row0 + ln, M - 1);                 // clamped A row
  const float* Arow = A + (size_t)am * lda + 2 * kh;
  const float* Wr0 = W + (size_t)imin(col0 +  0 + ln, N - 1) * ldw + 2 * kh;
  const float* Wr1 = W + (size_t)imin(col0 + 16 + ln, N - 1) * ldw + 2 * kh;
  const float* Wr2 = W + (size_t)imin(col0 + 32 + ln, N - 1) * ldw + 2 * kh;
  const float* Wr3 = W + (size_t)imin(col0 + 48 + ln, N - 1) * ldw + 2 * kh;

#pragma unroll 2
  for (int k = 0; k < K; k += 4) {
    const v2f a  = *(const v2f*)(Arow + k);   // A[m][k+2kh .. k+2kh+1]
    const v2f b0 = *(const v2f*)(Wr0 + k);    // B[k'][n] = W[n][k']
    const v2f b1 = *(const v2f*)(Wr1 + k);
    const v2f b2 = *(const v2f*)(Wr2 + k);
    const v2f b3 = *(const v2f*)(Wr3 + k);
    acc[0] = __builtin_amdgcn_wmma_f32_16x16x4_f32(false, a, false, b0, (short)0, acc[0], false, false);
    acc[1] = __builtin_amdgcn_wmma_f32_16x16x4_f32(false, a, false, b1, (short)0, acc[1], false, false);
    acc[2] = __builtin_amdgcn_wmma_f32_16x16x4_f32(false, a, false, b2, (short)0, acc[2], false, false);
    acc[3] = __builtin_amdgcn_wmma_f32_16x16x4_f32(false, a, false, b3, (short)0, acc[3], false, false);
  }

#pragma unroll
  for (int t = 0; t < 4; ++t) {
    const int n = col0 + t * 16 + ln;
    if (n >= N) continue;
    const float bv = bias ? bias[n] : 0.0f;
#pragma unroll
    for (int r = 0; r < 8; ++r) {
      const int m = row0 + r + 8 * kh;        // D: VGPR r -> M=r (lanes<16), M=r+8
      if (m < M) {
        float v = acc[t][r] + bv;
        if      (act == 1) v = silu_f(v);
        else if (act == 2) v = gelu_f(v);
        else if (act == 3) v = softplus_f(v);
        C[(size_t)m * ldc + n] = v;
      }
    }
  }
}

// --------------------- LayerNorm over channel-concat (dim=512) --------------
__global__ __launch_bounds__(256) void k_ln_c(
    const float* __restrict__ x1, const float* __restrict__ x2,
    const float* __restrict__ x3, const float* __restrict__ x4,
    const float* __restrict__ w, const float* __restrict__ bb,
    float* __restrict__ out)
{
  __shared__ float ssum[256], ssq[256];
  const int r = blockIdx.x;                  // 0..783  (b*196+l)
  const int t = threadIdx.x;
  const int c = t & 127;
  const float* p0 = (t < 128) ? x1 : x2;
  const float* p1 = (t < 128) ? x3 : x4;
  const float v0 = p0[(size_t)r * 128 + c];
  const float v1 = p1[(size_t)r * 128 + c];
  ssum[t] = v0 + v1; ssq[t] = v0 * v0 + v1 * v1;
  __syncthreads();
  for (int s = 128; s > 0; s >>= 1) {
    if (t < s) { ssum[t] += ssum[t + s]; ssq[t] += ssq[t + s]; }
    __syncthreads();
  }
  const float mean = ssum[0] * (1.0f / 512.0f);
  const float var  = ssq[0] * (1.0f / 512.0f) - mean * mean;
  const float rstd = rsqrtf(var + LN_EPS);
  const int j0 = t, j1 = t + 256;
  out[(size_t)r * 512 + j0] = (v0 - mean) * rstd * w[j0] + bb[j0];
  out[(size_t)r * 512 + j1] = (v1 - mean) * rstd * w[j1] + bb[j1];
}

// --------------------- LayerNorm over seq-concat (dim=128) ------------------
__global__ __launch_bounds__(128) void k_ln_s(
    const float* __restrict__ x1, const float* __restrict__ x2,
    const float* __restrict__ x3, const float* __restrict__ x4,
    const float* __restrict__ w, const float* __restrict__ bb,
    float* __restrict__ out)
{
  __shared__ float ssum[128], ssq[128];
  const int r = blockIdx.x;                  // 0..3135 : b*784 + i*196 + l
  const int b = r / 784, rem = r % 784, i = rem / 196, l = rem % 196;
  const float* xi = (i == 0) ? x1 : (i == 1) ? x2 : (i == 2) ? x3 : x4;
  const int t = threadIdx.x;
  const float v = xi[(size_t)(b * 196 + l) * 128 + t];
  ssum[t] = v; ssq[t] = v * v;
  __syncthreads();
  for (int s = 64; s > 0; s >>= 1) {
    if (t < s) { ssum[t] += ssum[t + s]; ssq[t] += ssq[t + s]; }
    __syncthreads();
  }
  const float mean = ssum[0] * (1.0f / 128.0f);
  const float var  = ssq[0] * (1.0f / 128.0f) - mean * mean;
  const float rstd = rsqrtf(var + LN_EPS);
  out[(size_t)r * 128 + t] = (v - mean) * rstd * w[t] + bb[t];
}

// --------------------- causal depthwise conv (K=4) + SiLU -------------------
// reads xs half of xz (stride 2*di), writes (B*L, di)
__global__ __launch_bounds__(256) void k_conv_silu(
    const float* __restrict__ xz, const float* __restrict__ w,
    const float* __restrict__ bias, float* __restrict__ out,
    int Bn, int L, int di)
{
  const int idx = blockIdx.x * 256 + threadIdx.x;
  if (idx >= Bn * L * di) return;
  const int c = idx % di;
  const int l = (idx / di) % L;
  const int b = idx / (di * L);
  const int zld = 2 * di;
  const float* base = xz + (size_t)(b * L) * zld + c;
  float acc = bias[c];
#pragma unroll
  for (int k = 0; k < 4; ++k) {
    const int ls = l - 3 + k;
    if (ls >= 0) acc += w[c * 4 + k] * base[(size_t)ls * zld];
  }
  out[idx] = silu_f(acc);
}

// --------------------- selective scan (fused D, silu(z) gate) ---------------
__global__ __launch_bounds__(256) void k_scan(
    const float* __restrict__ dt, const float* __restrict__ u,
    const float* __restrict__ xdbl, int xld, int boff, int coff,
    const float* __restrict__ A_log, const float* __restrict__ Dp,
    const float* __restrict__ xz, int zld, int zoff,
    float* __restrict__ y, int L, int di)
{
  const int t = blockIdx.x * 256 + threadIdx.x;
  const int b = t / di, d = t % di;
  float a[16], h[16];
#pragma unroll
  for (int n = 0; n < 16; ++n) { a[n] = -expf(A_log[d * 16 + n]); h[n] = 0.0f; }
  const float Dd = Dp[d];
  for (int l = 0; l < L; ++l) {
    const size_t r = (size_t)b * L + l;
    const float dtv = dt[r * di + d];
    const float uv  = u[r * di + d];
    const float dtu = dtv * uv;
    const float* Bp = xdbl + r * xld + boff;
    const float* Cp = xdbl + r * xld + coff;
    float yv = 0.0f;
#pragma unroll
    for (int n = 0; n < 16; ++n) {
      h[n] = expf(dtv * a[n]) * h[n] + dtu * Bp[n];
      yv += h[n] * Cp[n];
    }
    const float zv = xz[r * zld + zoff + d];
    y[r * di + d] = (yv + uv * Dd) * silu_f(zv);
  }
}

// m1cp[b, i*196+l, c] = m1c[b*196+l, i*128+c] + xi[b,l,c]
__global__ __launch_bounds__(256) void k_m1cp(
    const float* __restrict__ m1c,
    const float* __restrict__ x1, const float* __restrict__ x2,
    const float* __restrict__ x3, const float* __restrict__ x4,
    float* __restrict__ out)
{
  const int idx = blockIdx.x * 256 + threadIdx.x;
  if (idx >= 4 * 784 * 128) return;
  const int c = idx % 128, ri = idx / 128;
  const int b = ri / 784, rem = ri % 784, i = rem / 196, l = rem % 196;
  const float* xi = (i == 0) ? x1 : (i == 1) ? x2 : (i == 2) ? x3 : x4;
  out[idx] = m1c[(size_t)(b * 196 + l) * 512 + i * 128 + c]
           + xi[(size_t)(b * 196 + l) * 128 + c];
}

// in-place residual: m1s += s1 (original, pre-layernorm concat)
__global__ __launch_bounds__(256) void k_m1s_res(
    float* __restrict__ m1s,
    const float* __restrict__ x1, const float* __restrict__ x2,
    const float* __restrict__ x3, const float* __restrict__ x4)
{
  const int idx = blockIdx.x * 256 + threadIdx.x;
  if (idx >= 4 * 784 * 128) return;
  const int c = idx % 128, ri = idx / 128;
  const int b = ri / 784, rem = ri % 784, i = rem / 196, l = rem % 196;
  const float* xi = (i == 0) ? x1 : (i == 1) ? x2 : (i == 2) ? x3 : x4;
  m1s[idx] += xi[(size_t)(b * 196 + l) * 128 + c];
}

// sigmoid fusion gate + weighted sum
__global__ __launch_bounds__(128) void k_fusion(
    const float* __restrict__ m1cp, const float* __restrict__ m1s,
    const float* __restrict__ fw, const float* __restrict__ fb,
    float* __restrict__ fused)
{
  __shared__ float s0[128], s1[128];
  const int r = blockIdx.x, t = threadIdx.x;
  const float a = m1cp[(size_t)r * 128 + t];
  const float b = m1s[(size_t)r * 128 + t];
  s0[t] = fw[t] * a + fw[128 + t] * b;
  s1[t] = fw[256 + t] * a + fw[384 + t] * b;
  __syncthreads();
  for (int s = 64; s > 0; s >>= 1) {
    if (t < s) { s0[t] += s0[t + s]; s1[t] += s1[t + s]; }
    __syncthreads();
  }
  const float w0 = sigmoid_f(s0[0] + fb[0]);
  const float w1 = sigmoid_f(s1[0] + fb[1]);
  fused[(size_t)r * 128 + t] = w0 * a + w1 * b;
}

// a1[b*196+l, c] = gelu(fused[b*784+i*196+l, c]*dw_w[c] + dw_b[c])
__global__ __launch_bounds__(256) void k_dwgelu(
    const float* __restrict__ fused, const float* __restrict__ dw_w,
    const float* __restrict__ dw_b, float* __restrict__ a1, int scale_i)
{
  const int idx = blockIdx.x * 256 + threadIdx.x;
  if (idx >= 784 * 128) return;
  const int c = idx % 128, r = idx / 128;
  const int b = r / 196, l = r % 196;
  const float v = fused[(size_t)(b * 784 + scale_i * 196 + l) * 128 + c];
  a1[idx] = gelu_f(v * dw_w[c] + dw_b[c]);
}

// per-channel batchnorm stats -> affine (alpha,beta); stats at 14x14 base ==
// stats of the nearest-upsampled tensor, so this is exact.
__global__ __launch_bounds__(256) void k_bn_stats(
    const float* __restrict__ x, int rows, int C,
    const float* __restrict__ g, const float* __restrict__ bb,
    float* __restrict__ alpha, float* __restrict__ beta)
{
  __shared__ float s[256], s2[256];
  const int c = blockIdx.x, t = threadIdx.x;
  float sum = 0.f, sq = 0.f;
  for (int r = t; r < rows; r += 256) {
    const float v = x[(size_t)r * C + c];
    sum += v; sq += v * v;
  }
  s[t] = sum; s2[t] = sq;
  __syncthreads();
  for (int st = 128; st > 0; st >>= 1) {
    if (t < st) { s[t] += s[t + st]; s2[t] += s2[t + st]; }
    __syncthreads();
  }
  if (t == 0) {
    const float mean = s[0] / (float)rows;
    const float var  = s2[0] / (float)rows - mean * mean;
    const float rstd = rsqrtf(var + LN_EPS);
    alpha[c] = g[c] * rstd;
    beta[c]  = bb[c] - mean * rstd * g[c];
  }
}

__global__ __launch_bounds__(256) void k_bn_apply(
    float* __restrict__ x, int total, int C,
    const float* __restrict__ alpha, const float* __restrict__ beta)
{
  const int idx = blockIdx.x * 256 + threadIdx.x;
  if (idx >= total) return;
  const int c = idx % C;
  x[idx] = x[idx] * alpha[c] + beta[c];
}

// nearest-neighbor expand: src (b*196+l, C) row-major -> dst (B,C,14s,14s)
__global__ __launch_bounds__(256) void k_upsample(
    const float* __restrict__ src, float* __restrict__ dst, int Cc, int s)
{
  const int HW = 14 * s;
  const int total = 4 * Cc * HW * HW;
  const int idx = blockIdx.x * 256 + threadIdx.x;
  if (idx >= total) return;
  const int x = idx % HW;
  const int y = (idx / HW) % HW;
  const int c = (idx / (HW * HW)) % Cc;
  const int b = idx / (HW * HW * Cc);
  const int l = (y / s) * 14 + (x / s);
  dst[idx] = src[(size_t)(b * 196 + l) * Cc + c];
}

// ---------------------------------------------------------------------------
extern "C" void kernel_launch(void* const* d_in, const int* in_sizes, int n_in,
                              void* d_out, int out_size, void* d_ws, size_t ws_size,
                              hipStream_t stream)
{
  (void)in_sizes; (void)n_in; (void)out_size; (void)ws_size;
  const float* x1 = (const float*)d_in[0];
  const float* x2 = (const float*)d_in[1];
  const float* x3 = (const float*)d_in[2];
  const float* x4 = (const float*)d_in[3];
  const float* cnorm_w = (const float*)d_in[4];
  const float* cnorm_b = (const float*)d_in[5];
  const float* snorm_w = (const float*)d_in[6];
  const float* snorm_b = (const float*)d_in[7];
  // cm (dm=512, di=1024, dtr=32, N=16)
  const float* cm_inw  = (const float*)d_in[8];   // (2048,512)
  const float* cm_cw   = (const float*)d_in[9];   // (1024,4)
  const float* cm_cb   = (const float*)d_in[10];
  const float* cm_xw   = (const float*)d_in[11];  // (64,1024)
  const float* cm_dtw  = (const float*)d_in[12];  // (1024,32)
  const float* cm_dtb  = (const float*)d_in[13];
  const float* cm_Alog = (const float*)d_in[14];  // (1024,16)
  const float* cm_D    = (const float*)d_in[15];
  const float* cm_ow   = (const float*)d_in[16];  // (512,1024)
  // sm (dm=128, di=256, dtr=8, N=16)
  const float* sm_inw  = (const float*)d_in[17];  // (512,128)
  const float* sm_cw   = (const float*)d_in[18];
  const float* sm_cb   = (const float*)d_in[19];
  const float* sm_xw   = (const float*)d_in[20];  // (40,256)
  const float* sm_dtw  = (const float*)d_in[21];  // (256,8)
  const float* sm_dtb  = (const float*)d_in[22];
  const float* sm_Alog = (const float*)d_in[23];
  const float* sm_D    = (const float*)d_in[24];
  const float* sm_ow   = (const float*)d_in[25];  // (128,256)
  const float* fusion_w = (const float*)d_in[26]; // (2,256)
  const float* fusion_b = (const float*)d_in[27];

  float* out = (float*)d_out;

  // ---- workspace layout ----
  char* wp = (char*)d_ws;
  auto alloc = [&](size_t nflt) -> float* {
    float* q = (float*)wp;
    wp += (((nflt * sizeof(float)) + 255) & ~(size_t)255);
    return q;
  };
  const int MC = 784, MS = 3136;
  float* xc_ln  = alloc((size_t)MC * 512);
  float* xs_ln  = alloc((size_t)MS * 128);
  float* xz_c   = alloc((size_t)MC * 2048);
  float* conv_c = alloc((size_t)MC * 1024);
  float* xdbl_c = alloc((size_t)MC * 64);
  float* dt_c   = alloc((size_t)MC * 1024);
  float* y_c    = alloc((size_t)MC * 1024);
  float* m1c    = alloc((size_t)MC * 512);
  float* m1cp   = alloc((size_t)MS * 128);
  float* xz_s   = alloc((size_t)MS * 512);
  float* conv_s = alloc((size_t)MS * 256);
  float* xdbl_s = alloc((size_t)MS * 40);
  float* dt_s   = alloc((size_t)MS * 256);
  float* y_s    = alloc((size_t)MS * 256);
  float* m1s    = alloc((size_t)MS * 128);
  float* fusedb = alloc((size_t)MS * 128);
  float* a1     = alloc((size_t)MC * 128);
  float* a2     = alloc((size_t)MC * 512);
  float* a3     = alloc((size_t)MC * 256);
  float* alpha  = alloc(512);
  float* beta   = alloc(512);

  // ---- layernorms (with concatenation) ----
  k_ln_c<<<MC, 256, 0, stream>>>(x1, x2, x3, x4, cnorm_w, cnorm_b, xc_ln);
  k_ln_s<<<MS, 128, 0, stream>>>(x1, x2, x3, x4, snorm_w, snorm_b, xs_ln);

  // ---- channel mamba ----
  k_gemm<<<dim3(32, 7), 256, 0, stream>>>(xc_ln, 512, cm_inw, 512, xz_c, 2048,
                                          MC, 2048, 512, nullptr, 0);
  k_conv_silu<<<(4*196*1024 + 255)/256, 256, 0, stream>>>(xz_c, cm_cw, cm_cb,
                                                          conv_c, 4, 196, 1024);
  k_gemm<<<dim3(1, 7), 256, 0, stream>>>(conv_c, 1024, cm_xw, 1024, xdbl_c, 64,
                                         MC, 64, 1024, nullptr, 0);
  k_gemm<<<dim3(16, 7), 256, 0, stream>>>(xdbl_c, 64, cm_dtw, 32, dt_c, 1024,
                                          MC, 1024, 32, cm_dtb, 3 /*softplus*/);
  k_scan<<<16, 256, 0, stream>>>(dt_c, conv_c, xdbl_c, 64, 32, 48,
                                 cm_Alog, cm_D, xz_c, 2048, 1024,
                                 y_c, 196, 1024);
  k_gemm<<<dim3(8, 7), 256, 0, stream>>>(y_c, 1024, cm_ow, 1024, m1c, 512,
                                         MC, 512, 1024, nullptr, 0);
  k_m1cp<<<(4*784*128 + 255)/256, 256, 0, stream>>>(m1c, x1, x2, x3, x4, m1cp);

  // ---- spatial mamba ----
  k_gemm<<<dim3(8, 25), 256, 0, stream>>>(xs_ln, 128, sm_inw, 128, xz_s, 512,
                                          MS, 512, 128, nullptr, 0);
  k_conv_silu<<<(4*784*256 + 255)/256, 256, 0, stream>>>(xz_s, sm_cw, sm_cb,
                                                         conv_s, 4, 784, 256);
  k_gemm<<<dim3(1, 25), 256, 0, stream>>>(conv_s, 256, sm_xw, 256, xdbl_s, 40,
                                          MS, 40, 256, nullptr, 0);
  k_gemm<<<dim3(4, 25), 256, 0, stream>>>(xdbl_s, 40, sm_dtw, 8, dt_s, 256,
                                          MS, 256, 8, sm_dtb, 3 /*softplus*/);
  k_scan<<<4, 256, 0, stream>>>(dt_s, conv_s, xdbl_s, 40, 8, 24,
                                sm_Alog, sm_D, xz_s, 512, 256,
                                y_s, 784, 256);
  k_gemm<<<dim3(2, 25), 256, 0, stream>>>(y_s, 256, sm_ow, 256, m1s, 128,
                                          MS, 128, 256, nullptr, 0);
  k_m1s_res<<<(4*784*128 + 255)/256, 256, 0, stream>>>(m1s, x1, x2, x3, x4);

  // ---- fusion gate ----
  k_fusion<<<MS, 128, 0, stream>>>(m1cp, m1s, fusion_w, fusion_b, fusedb);

  // ---- reconstruct heads (computed at 14x14 base, expanded at the end) ----
  const int scales[4] = {16, 8, 4, 2};
  const int couts[4]  = {64, 64, 128, 256};
  size_t out_off = 0;
  for (int i = 0; i < 4; ++i) {
    const int base = 28 + 12 * i;
    const float* dw_w  = (const float*)d_in[base + 0];
    const float* dw_b  = (const float*)d_in[base + 1];
    const float* bn1_g = (const float*)d_in[base + 2];
    const float* bn1_b = (const float*)d_in[base + 3];
    const float* pw1_w = (const float*)d_in[base + 4];  // (512,128)
    const float* pw1_b = (const float*)d_in[base + 5];
    const float* bn2_g = (const float*)d_in[base + 6];
    const float* bn2_b = (const float*)d_in[base + 7];
    const float* pw2_w = (const float*)d_in[base + 8];  // (cout,512)
    const float* pw2_b = (const float*)d_in[base + 9];
    const float* bn3_g = (const float*)d_in[base + 10];
    const float* bn3_b = (const float*)d_in[base + 11];
    const int s = scales[i], cout = couts[i];

    k_dwgelu<<<(MC*128 + 255)/256, 256, 0, stream>>>(fusedb, dw_w, dw_b, a1, i);
    k_bn_stats<<<128, 256, 0, stream>>>(a1, MC, 128, bn1_g, bn1_b, alpha, beta);
    k_bn_apply<<<(MC*128 + 255)/256, 256, 0, stream>>>(a1, MC*128, 128, alpha, beta);

    k_gemm<<<dim3(8, 7), 256, 0, stream>>>(a1, 128, pw1_w, 128, a2, 512,
                                           MC, 512, 128, pw1_b, 2 /*gelu*/);
    k_bn_stats<<<512, 256, 0, stream>>>(a2, MC, 512, bn2_g, bn2_b, alpha, beta);
    k_bn_apply<<<(MC*512 + 255)/256, 256, 0, stream>>>(a2, MC*512, 512, alpha, beta);

    k_gemm<<<dim3((cout + 63)/64, 7), 256, 0, stream>>>(a2, 512, pw2_w, 512, a3, cout,
                                                        MC, cout, 512, pw2_b, 2 /*gelu*/);
    k_bn_stats<<<cout, 256, 0, stream>>>(a3, MC, cout, bn3_g, bn3_b, alpha, beta);
    k_bn_apply<<<(MC*cout + 255)/256, 256, 0, stream>>>(a3, MC*cout, cout, alpha, beta);

    const int total = 4 * cout * (14*s) * (14*s);
    k_upsample<<<(total + 255)/256, 256, 0, stream>>>(a3, out + out_off, cout, s);
    out_off += (size_t)total;
  }
}